// UNet18_40k_SWIN_pred_bigtail_average_26388279067310
// MI455X (gfx1250) — compile-verified
//
#include <hip/hip_runtime.h>
#include <math.h>

#define KNB 19

typedef __attribute__((ext_vector_type(2))) float v2f;
typedef __attribute__((ext_vector_type(8))) float v8f;

// ---------------------------------------------------------------------------
// Kernels
// ---------------------------------------------------------------------------

// Y[N x Cout] = A[N x Cin] * W[Cin x Cout] (+ bias), fp32 WMMA 16x16x4.
// REQUIRES Cin % 4 == 0 (call sites guarantee this; init conv is zero-padded).
// Out-of-range rows/cols are index-CLAMPED, not zeroed: garbage only lands in
// D elements that are never stored, so the inner loop is fully branch-free.
__global__ void k_gemm_wmma(const float* __restrict__ A, const float* __restrict__ W,
                            const float* __restrict__ bias, float* __restrict__ Y,
                            int N, int Cin, int Cout) {
  const int wave = threadIdx.x >> 5;
  const int lane = threadIdx.x & 31;
  const int m0 = (blockIdx.x * 4 + wave) << 4;
  const int n0 = blockIdx.y << 4;
  if (m0 >= N) return;                       // uniform per wave
  const int half = lane >> 4;                // 0: lanes 0-15, 1: lanes 16-31
  const int idx  = lane & 15;
  const int m = min(m0 + idx, N - 1);        // clamped (safe, see above)
  const int n = min(n0 + idx, Cout - 1);     // clamped
  const float* __restrict__ arow = A + m * Cin;
  const float* __restrict__ wcol = W + n;
  v8f acc = {0.f, 0.f, 0.f, 0.f, 0.f, 0.f, 0.f, 0.f};
  for (int k0 = 0; k0 < Cin; k0 += 4) {
    const int ka = k0 + 2 * half;            // this lane's K pair
    // A frag: two contiguous K values -> single 64-bit load (8B aligned)
    const v2f af = *(const v2f*)(arow + ka);
    v2f bf;
    bf.x = wcol[ka * Cout];
    bf.y = wcol[(ka + 1) * Cout];
    acc = __builtin_amdgcn_wmma_f32_16x16x4_f32(false, af, false, bf,
                                                (short)0, acc, false, false);
  }
  const int nn = n0 + idx;
  if (nn >= Cout) return;
  const float bv = bias ? bias[nn] : 0.f;
#pragma unroll
  for (int j = 0; j < 8; ++j) {
    const int row = m0 + j + 8 * half;       // C/D layout: VGPR j -> M = j + 8*half
    if (row < N) Y[row * Cout + nn] = acc[j] + bv;
  }
}

// Pad x (N x 3) -> (N x 4) with zero in channel 3.
__global__ void k_pad4(const float* __restrict__ X, float* __restrict__ Y, int N) {
  const int i = blockIdx.x * blockDim.x + threadIdx.x;
  if (i >= N * 4) return;
  const int n = i >> 2, c = i & 3;
  Y[i] = (c < 3) ? X[n * 3 + c] : 0.f;
}

// Pad W (3 x Cout) -> (4 x Cout) with zero row 3.
__global__ void k_padW(const float* __restrict__ W, float* __restrict__ Y, int Cout) {
  const int i = blockIdx.x * blockDim.x + threadIdx.x;
  if (i >= 4 * Cout) return;
  Y[i] = (i < 3 * Cout) ? W[i] : 0.f;
}

// Window attention: one wave32 per node, channels striped across lanes.
// Branchless channel guards: clamped indices + zeroed q for inactive lanes.
__global__ void k_attn(const float* __restrict__ Q, const float* __restrict__ Km,
                       const float* __restrict__ Vm, const int* __restrict__ win,
                       float* __restrict__ O, int N, int C, float scale) {
  const int node = blockIdx.x * (blockDim.x >> 5) + (threadIdx.x >> 5);
  if (node >= N) return;
  const int lane = threadIdx.x & 31;
  const int nt = (C + 31) >> 5;              // <= 4 (C <= 128)
  float q[4];
  int cc[4];
#pragma unroll
  for (int t = 0; t < 4; ++t) {
    const int c = lane + t * 32;
    cc[t] = min(c, C - 1);
    q[t] = (t < nt && c < C) ? Q[node * C + cc[t]] : 0.f;  // 0 for pad lanes
  }
  int nb[KNB];
#pragma unroll
  for (int k = 0; k < KNB; ++k) nb[k] = win[node * KNB + k];
  float s[KNB];
#pragma unroll
  for (int k = 0; k < KNB; ++k) {
    float p = 0.f;
    const float* __restrict__ kr = Km + nb[k] * C;
    for (int t = 0; t < nt; ++t) p += q[t] * kr[cc[t]];    // q==0 masks pads
    for (int off = 16; off >= 1; off >>= 1) p += __shfl_xor(p, off, 32);
    s[k] = p * scale;
  }
  float mx = s[0];
#pragma unroll
  for (int k = 1; k < KNB; ++k) mx = fmaxf(mx, s[k]);
  float sum = 0.f;
#pragma unroll
  for (int k = 0; k < KNB; ++k) { s[k] = expf(s[k] - mx); sum += s[k]; }
  const float inv = 1.f / sum;
  for (int t = 0; t < nt; ++t) {
    const int c = lane + t * 32;
    float o = 0.f;
#pragma unroll
    for (int k = 0; k < KNB; ++k) o += s[k] * Vm[nb[k] * C + cc[t]];
    if (c < C) O[node * C + c] = o * inv;    // only store predicated
  }
}

// Per-channel mean/var over N (biased var, matches jnp.var). One block per channel.
__global__ void k_bn_stats(const float* __restrict__ X, float* __restrict__ stats,
                           int N, int C) {
  __shared__ float sh[512];
  const int c = blockIdx.x;
  float s = 0.f, ss = 0.f;
  for (int i = threadIdx.x; i < N; i += blockDim.x) {
    const float v = X[i * C + c];
    s += v; ss += v * v;
  }
  sh[threadIdx.x] = s;
  sh[256 + threadIdx.x] = ss;
  __syncthreads();
  for (int st = 128; st >= 1; st >>= 1) {
    if ((int)threadIdx.x < st) {
      sh[threadIdx.x]       += sh[threadIdx.x + st];
      sh[256 + threadIdx.x] += sh[256 + threadIdx.x + st];
    }
    __syncthreads();
  }
  if (threadIdx.x == 0) {
    const float m = sh[0] / (float)N;
    stats[c]     = m;
    stats[C + c] = sh[256] / (float)N - m * m;
  }
}

__global__ void k_bn_apply(const float* __restrict__ X, const float* __restrict__ stats,
                           const float* __restrict__ g, const float* __restrict__ b,
                           float* __restrict__ Y, int N, int C, int lrelu) {
  const int i = blockIdx.x * blockDim.x + threadIdx.x;
  if (i >= N * C) return;
  const int c = i % C;
  float y = g[c] * (X[i] - stats[c]) * rsqrtf(stats[C + c] + 1e-5f) + b[c];
  if (lrelu) y = (y >= 0.f) ? y : 0.2f * y;
  Y[i] = y;
}

__global__ void k_add(const float* __restrict__ A, const float* __restrict__ B,
                      float* __restrict__ Y, int n) {
  const int i = blockIdx.x * blockDim.x + threadIdx.x;
  if (i < n) Y[i] = A[i] + B[i];
}

// Y[m,c] = mean over 7 of X[po[m,j], c]
__global__ void k_pool_mean(const float* __restrict__ X, const int* __restrict__ po,
                            float* __restrict__ Y, int M, int C) {
  const int i = blockIdx.x * blockDim.x + threadIdx.x;
  if (i >= M * C) return;
  const int m = i / C, c = i % C;
  float s = 0.f;
#pragma unroll
  for (int j = 0; j < 7; ++j) s += X[po[m * 7 + j] * C + c];
  Y[i] = s * (1.f / 7.f);
}

// rows [Ns,Nb): Y[Ns+r] = 0.5*(Y[didx[r,0]] + Y[didx[r,1]]) (reads rows < Ns only)
__global__ void k_up_tail(float* __restrict__ Y, const int* __restrict__ didx,
                          int Ns, int Nb, int C) {
  const int i = blockIdx.x * blockDim.x + threadIdx.x;
  const int total = (Nb - Ns) * C;
  if (i >= total) return;
  const int r = i / C, c = i % C;
  const int i0 = didx[r * 2], i1 = didx[r * 2 + 1];
  Y[(Ns + r) * C + c] = 0.5f * (Y[i0 * C + c] + Y[i1 * C + c]);
}

// channel concat: Y[n, 0:Ca) = A[n], Y[n, Ca:Ca+Cb) = B[n]
__global__ void k_concat(const float* __restrict__ A, const float* __restrict__ B,
                         float* __restrict__ Y, int N, int Ca, int Cb) {
  const int C = Ca + Cb;
  const int i = blockIdx.x * blockDim.x + threadIdx.x;
  if (i >= N * C) return;
  const int n = i / C, c = i % C;
  Y[i] = (c < Ca) ? A[n * Ca + c] : B[n * Cb + (c - Ca)];
}

// ---------------------------------------------------------------------------
// Host-side parameter structs + flattening-order cursor
// ---------------------------------------------------------------------------

struct AttnP  { const float *wq, *wk, *wv, *wo, *bo; };
struct BlockP { AttnP a1, a2; const float *g1, *b1, *g2, *b2; };
struct ConvP  { const float *w, *b; };
struct PmP    { const float *w, *b, *g, *bt; };

struct Cursor { void* const* d; int i; };
static inline const float* nf(Cursor& c) { return (const float*)c.d[c.i++]; }
static inline const int*   ni(Cursor& c) { return (const int*)c.d[c.i++]; }

static AttnP getAttn(Cursor& c, bool srt) {
  AttnP a;
  if (srt) { a.bo = nf(c); a.wk = nf(c); a.wo = nf(c); a.wq = nf(c); a.wv = nf(c); }
  else     { a.wq = nf(c); a.wk = nf(c); a.wv = nf(c); a.wo = nf(c); a.bo = nf(c); }
  return a;
}
static BlockP getBlock(Cursor& c, bool srt) {
  BlockP b;
  if (srt) { b.a1 = getAttn(c, 1); b.a2 = getAttn(c, 1);
             b.b1 = nf(c); b.b2 = nf(c); b.g1 = nf(c); b.g2 = nf(c); }
  else     { b.a1 = getAttn(c, 0); b.g1 = nf(c); b.b1 = nf(c);
             b.a2 = getAttn(c, 0); b.g2 = nf(c); b.b2 = nf(c); }
  return b;
}
static ConvP getConv(Cursor& c, bool srt) {
  ConvP p;
  if (srt) { p.b = nf(c); p.w = nf(c); } else { p.w = nf(c); p.b = nf(c); }
  return p;
}
static PmP getPm(Cursor& c, bool srt) {
  PmP p;
  if (srt) { p.b = nf(c); p.bt = nf(c); p.g = nf(c); p.w = nf(c); }
  else     { p.w = nf(c); p.b = nf(c); p.g = nf(c); p.bt = nf(c); }
  return p;
}

// ---------------------------------------------------------------------------
// Host-side layer drivers
// ---------------------------------------------------------------------------

static void gemm(const float* A, const float* W, const float* bias, float* Y,
                 int N, int Cin, int Cout, hipStream_t s) {
  dim3 g((N + 63) / 64, (Cout + 15) / 16);
  k_gemm_wmma<<<g, 128, 0, s>>>(A, W, bias, Y, N, Cin, Cout);
}

static void bn(const float* X, const float* g, const float* b, float* Y,
               int N, int C, int lrelu, float* stats, hipStream_t s) {
  k_bn_stats<<<C, 256, 0, s>>>(X, stats, N, C);
  const int tot = N * C;
  k_bn_apply<<<(tot + 255) / 256, 256, 0, s>>>(X, stats, g, b, Y, N, C, lrelu);
}

static void attn_layer(const float* X, const AttnP& a, const int* win,
                       int N, int Cin, int Cout,
                       float* Q, float* Kt, float* Vt, float* O, float* Yout,
                       hipStream_t s) {
  gemm(X, a.wq, nullptr, Q,  N, Cin, Cin, s);
  gemm(X, a.wk, nullptr, Kt, N, Cin, Cin, s);  // gather commutes with linear map
  gemm(X, a.wv, nullptr, Vt, N, Cin, Cin, s);
  const float scale = 1.f / sqrtf((float)Cin);
  k_attn<<<(N + 7) / 8, 256, 0, s>>>(Q, Kt, Vt, win, O, N, Cin, scale);
  gemm(O, a.wo, a.bo, Yout, N, Cin, Cout, s);
}

static void run_block(const BlockP& bp, const float* X, const int* win,
                      int N, int Cin, int Cout,
                      float* t0, float* t1, float* t2, float* t3,
                      float* stats, float* Yout, hipStream_t s) {
  // attn1: Q=t0 K=t1 V=t2 O=t3, y1 -> t0
  attn_layer(X, bp.a1, win, N, Cin, Cout, t0, t1, t2, t3, t0, s);
  bn(t0, bp.g1, bp.b1, t1, N, Cout, 1, stats, s);            // H = t1
  // attn2 on H: Q=t2 K=t3 V=t0 O=t1 (H consumed), y2 -> t2
  attn_layer(t1, bp.a2, win, N, Cout, Cout, t2, t3, t0, t1, t2, s);
  bn(t2, bp.g2, bp.b2, Yout, N, Cout, 1, stats, s);
}

static void run_pmerge(const PmP& pp, const float* Xsum, const int* po,
                       int Nout, int Cin, int Cout,
                       float* tA, float* tB, float* stats, float* Yout,
                       hipStream_t s) {
  k_pool_mean<<<(Nout * Cin + 255) / 256, 256, 0, s>>>(Xsum, po, tA, Nout, Cin);
  gemm(tA, pp.w, pp.b, tB, Nout, Cin, Cout, s);
  bn(tB, pp.g, pp.bt, Yout, Nout, Cout, 0, stats, s);
}

// ---------------------------------------------------------------------------
// Entry point
// ---------------------------------------------------------------------------

extern "C" void kernel_launch(void* const* d_in, const int* in_sizes, int n_in,
                              void* d_out, int out_size, void* d_ws, size_t ws_size,
                              hipStream_t stream) {
  (void)n_in; (void)out_size; (void)ws_size;

  // ---- collect inputs under either flattening convention ----
  const float* x;
  const int *wins[5], *pools[4], *upsI[4];
  ConvP initp, outc, us[4];
  BlockP down[5], trans[4], up[4];
  PmP pm[4];

  const bool insertion = (in_sizes[0] > 64);  // x first (122886) vs bias first (8)
  Cursor c{d_in, 0};
  if (insertion) {
    x = nf(c);
    for (int i = 0; i < 5; ++i) wins[i] = ni(c);
    for (int i = 0; i < 4; ++i) pools[i] = ni(c);
    for (int i = 0; i < 4; ++i) upsI[i] = ni(c);
    initp = getConv(c, 0);
    down[0] = getBlock(c, 0);
    pm[0] = getPm(c, 0); down[1] = getBlock(c, 0);
    pm[1] = getPm(c, 0); down[2] = getBlock(c, 0);
    pm[2] = getPm(c, 0); down[3] = getBlock(c, 0);
    pm[3] = getPm(c, 0); down[4] = getBlock(c, 0);
    for (int j = 0; j < 4; ++j) {
      us[j] = getConv(c, 0); trans[j] = getBlock(c, 0); up[j] = getBlock(c, 0);
    }
    outc = getConv(c, 0);
  } else {  // jax tree_leaves: sorted dict keys at every level
    for (int i = 0; i < 5; ++i) down[i] = getBlock(c, 1);
    initp = getConv(c, 1);
    outc  = getConv(c, 1);
    for (int i = 0; i < 4; ++i) pm[i] = getPm(c, 1);
    for (int i = 0; i < 4; ++i) trans[i] = getBlock(c, 1);
    for (int i = 0; i < 4; ++i) up[i] = getBlock(c, 1);
    for (int i = 0; i < 4; ++i) us[i] = getConv(c, 1);
    for (int i = 0; i < 4; ++i) pools[i] = ni(c);
    for (int i = 0; i < 4; ++i) upsI[i] = ni(c);
    for (int i = 0; i < 5; ++i) wins[i] = ni(c);
    x = nf(c);
  }

  // ---- workspace carve (floats, 256B aligned) ----
  char* base = (char*)d_ws;
  size_t off = 0;
  auto alloc = [&](size_t nflt) -> float* {
    float* p = (float*)(base + off);
    off += ((nflt * sizeof(float) + 255) / 256) * 256;
    return p;
  };

  const int NSa[5] = {40962, 10242, 2562, 642, 162};
  const int Cs[5]  = {8, 16, 32, 64, 64};

  float* xs[5];  float* xs1[5];
  for (int l = 0; l < 5; ++l) { xs[l] = alloc((size_t)NSa[l] * Cs[l]);
                                xs1[l] = alloc((size_t)NSa[l] * Cs[l]); }
  float* t0 = alloc((size_t)40962 * 128);
  float* t1 = alloc((size_t)40962 * 128);
  float* t2 = alloc((size_t)40962 * 128);
  float* t3 = alloc((size_t)40962 * 128);
  float* cat = alloc((size_t)40962 * 128);
  float* ua  = alloc((size_t)40962 * 64);
  float* usb = alloc((size_t)40962 * 64);
  float* tb  = alloc((size_t)40962 * 64);
  float* stats = alloc(256);
  float* xpad  = alloc((size_t)40962 * 4);
  float* wpad  = alloc(32);

  // ---- encoder ----
  // init conv: zero-pad Cin 3 -> 4 so every GEMM has Cin % 4 == 0
  k_pad4<<<(NSa[0] * 4 + 255) / 256, 256, 0, stream>>>(x, xpad, NSa[0]);
  k_padW<<<1, 64, 0, stream>>>(initp.w, wpad, 8);
  gemm(xpad, wpad, initp.b, xs[0], NSa[0], 4, 8, stream);           // x1
  for (int l = 0; l < 5; ++l) {
    run_block(down[l], xs[l], wins[l], NSa[l], Cs[l], Cs[l],
              t0, t1, t2, t3, stats, xs1[l], stream);               // x{l+1}_1
    if (l < 4) {
      const int n = NSa[l] * Cs[l];
      k_add<<<(n + 255) / 256, 256, 0, stream>>>(xs[l], xs1[l], cat, n);
      run_pmerge(pm[l], cat, pools[l], NSa[l + 1], Cs[l], Cs[l + 1],
                 t0, t1, stats, xs[l + 1], stream);                 // x{l+2}
    }
  }

  // ---- decoder ----
  const float* uin = xs1[4];                        // x5_1 (162 x 64)
  const float* transX[4] = {xs1[3], xs1[2], xs1[1], xs1[0]};
  const int tCin[4]  = {64, 32, 16, 8};
  const int NsA[4]   = {162, 642, 2562, 10242};
  const int NbA[4]   = {642, 2562, 10242, 40962};
  const int winIx[4] = {3, 2, 1, 0};

  for (int j = 0; j < 4; ++j) {
    const int Ns = NsA[j], Nb = NbA[j];
    const int* win = wins[winIx[j]];
    // skip-branch transformer block
    run_block(trans[j], transX[j], win, Nb, tCin[j], 64,
              t0, t1, t2, t3, stats, tb, stream);
    // upsample branch: conv then 2-way-mean tail rows
    gemm(uin, us[j].w, us[j].b, usb, Ns, 64, 64, stream);
    const int tail = (Nb - Ns) * 64;
    k_up_tail<<<(tail + 255) / 256, 256, 0, stream>>>(usb, upsI[j], Ns, Nb, 64);
    // channel concat (128 ch) and fuse block
    const int cn = Nb * 128;
    k_concat<<<(cn + 255) / 256, 256, 0, stream>>>(usb, tb, cat, Nb, 64, 64);
    run_block(up[j], cat, win, Nb, 128, 64, t0, t1, t2, t3, stats, ua, stream);
    uin = ua;
  }

  // ---- output head: (40962 x 64) @ (64 x 3) -> d_out (B,N,3) node-major ----
  gemm(uin, outc.w, outc.b, (float*)d_out, NSa[0], 64, 3, stream);
}